// DirectVoxGO_77008763617691
// MI455X (gfx1250) — compile-verified
//
#include <hip/hip_runtime.h>
#include <math.h>

// Problem constants (from setup_inputs)
#define NRAYS  16384
#define NPTS   256
#define GS     160                   // grid size per dim (D = H = W)
#define GVOL   (GS * GS * GS)        // 4,096,000 voxels per channel
#define NCH    4                     // density + rgb, interleaved in packed grid

#define LOG2E      1.44269504088896340736f
// log(1/(1-1e-6) - 1)  (f32-rounded, like the reference)
#define ACT_SHIFT  (-13.8155095579637684f)

// ---------------------------------------------------------------------------
// Repack kernel: [4ch][D][H][W] planes (16 MB apart)  ->  [D][H][W][4ch] AoS.
// One-time 130 MB of traffic per launch; the 65 MB result stays resident in
// the 192 MB L2 and quarters the cacheline traffic of the gather phase.
// Source reads are non-temporal: the planar grids are never touched again.
// ---------------------------------------------------------------------------
__global__ __launch_bounds__(256) void dvgo_repack_kernel(
    const float* __restrict__ dgrid,   // [1,160,160,160]
    const float* __restrict__ cgrid,   // [3,160,160,160]
    float4* __restrict__ packed)       // [160*160*160] x float4
{
    const int lin = blockIdx.x * 256 + threadIdx.x;   // GVOL/256 = 16000 blocks exact
    if (lin >= GVOL) return;
    float4 v;
    v.x = __builtin_nontemporal_load(dgrid + lin);
    v.y = __builtin_nontemporal_load(cgrid + lin);
    v.z = __builtin_nontemporal_load(cgrid + GVOL + lin);
    v.w = __builtin_nontemporal_load(cgrid + 2 * GVOL + lin);
    packed[lin] = v;   // regular (RT) store: keep hot in L2 for the main kernel
}

// ---------------------------------------------------------------------------
// Main kernel over the packed AoS grid: 8 x global_load_b128 gathers/point.
// ---------------------------------------------------------------------------
__global__ __launch_bounds__(NPTS) void dvgo_render_packed_kernel(
    const float* __restrict__ origins,     // [R,3]
    const float* __restrict__ directions,  // [R,3]
    const float* __restrict__ lengths,     // [R,P]
    const float4* __restrict__ packed,     // [D][H][W] x (dens,r,g,b)
    float* __restrict__ out)               // density [R*P] ++ color [R*P*3]
{
    const int ray = blockIdx.x;
    const int p   = threadIdx.x;
    const int idx = ray * NPTS + p;

    // Uniform per-ray data -> scalar loads
    const float ox = origins[3 * ray + 0];
    const float oy = origins[3 * ray + 1];
    const float oz = origins[3 * ray + 2];
    const float dx = directions[3 * ray + 0];
    const float dy = directions[3 * ray + 1];
    const float dz = directions[3 * ray + 2];
    const float interval = sqrtf(fmaf(dx, dx, fmaf(dy, dy, dz * dz)));

    const float t = __builtin_nontemporal_load(lengths + idx);

    // World -> voxel coords: pos = ((p+1)/2) * (GS-1)
    const float S = 0.5f * (float)(GS - 1);
    const float fx = (fmaf(dx, t, ox) + 1.0f) * S;
    const float fy = (fmaf(dy, t, oy) + 1.0f) * S;
    const float fz = (fmaf(dz, t, oz) + 1.0f) * S;

    const float flx = floorf(fx), fly = floorf(fy), flz = floorf(fz);
    const float wx1 = fx - flx, wy1 = fy - fly, wz1 = fz - flz;
    const float wx0 = 1.0f - wx1, wy0 = 1.0f - wy1, wz0 = 1.0f - wz1;

    const int x0 = min(max((int)flx, 0), GS - 1);
    const int y0 = min(max((int)fly, 0), GS - 1);
    const int z0 = min(max((int)flz, 0), GS - 1);
    const int y1 = min(y0 + 1, GS - 1);
    const int z1 = min(z0 + 1, GS - 1);

    // Top-edge clamp folded into the x-weights (2 selects total):
    // if x0==159 both reference corners are voxel 159 with total weight wx0+wx1.
    const int   xb  = min(x0, GS - 2);
    const bool  xhi = (x0 != xb);
    const float wxa = xhi ? 0.0f : wx0;
    const float wxb = xhi ? (wx0 + wx1) : wx1;

    const int o00 = (z0 * GS + y0) * GS + xb;
    const int o01 = (z0 * GS + y1) * GS + xb;
    const int o10 = (z1 * GS + y0) * GS + xb;
    const int o11 = (z1 * GS + y1) * GS + xb;

    // 8 b128 gathers: each fetches all 4 channels of one corner (L2-resident).
    const float4 a0 = packed[o00], a1 = packed[o00 + 1];
    const float4 b0 = packed[o01], b1 = packed[o01 + 1];
    const float4 c0 = packed[o10], c1 = packed[o10 + 1];
    const float4 e0 = packed[o11], e1 = packed[o11 + 1];

    // Trilinear blend, 4 channels at once
    auto lx = [&](float v0, float v1) -> float {
        return fmaf(v0, wxa, v1 * wxb);
    };
    auto tri = [&](float pa0, float pa1, float pb0, float pb1,
                   float pc0, float pc1, float pe0, float pe1) -> float {
        const float q0 = fmaf(lx(pa0, pa1), wy0, lx(pb0, pb1) * wy1);
        const float q1 = fmaf(lx(pc0, pc1), wy0, lx(pe0, pe1) * wy1);
        return fmaf(q0, wz0, q1 * wz1);
    };

    const float raw_d = tri(a0.x, a1.x, b0.x, b1.x, c0.x, c1.x, e0.x, e1.x);
    const float raw_r = tri(a0.y, a1.y, b0.y, b1.y, c0.y, c1.y, e0.y, e1.y);
    const float raw_g = tri(a0.z, a1.z, b0.z, b1.z, c0.z, c1.z, e0.z, e1.z);
    const float raw_b = tri(a0.w, a1.w, b0.w, b1.w, c0.w, c1.w, e0.w, e1.w);

    // density = 1 - (1+e)^(-interval); v_exp/v_log are base-2, factors cancel:
    //   (1+e)^(-i) = exp2(-i * log2(1+e)),  e = exp2(raw*log2e + shift*log2e)
    const float e    = __builtin_amdgcn_exp2f(fmaf(raw_d, LOG2E, ACT_SHIFT * LOG2E));
    const float dens = 1.0f - __builtin_amdgcn_exp2f(-interval * __builtin_amdgcn_logf(1.0f + e));

    auto sigmoid = [](float x) -> float {
        return __builtin_amdgcn_rcpf(1.0f + __builtin_amdgcn_exp2f(-x * LOG2E));
    };
    const float col_r = sigmoid(raw_r);
    const float col_g = sigmoid(raw_g);
    const float col_b = sigmoid(raw_b);

    // Streaming (TH=NT) stores: don't evict the L2-resident packed grid.
    __builtin_nontemporal_store(dens, out + idx);
    float* __restrict__ cout = out + (size_t)NRAYS * NPTS + 3 * (size_t)idx;
    __builtin_nontemporal_store(col_r, cout + 0);
    __builtin_nontemporal_store(col_g, cout + 1);
    __builtin_nontemporal_store(col_b, cout + 2);
}

// ---------------------------------------------------------------------------
// Fallback (workspace too small): round-1 style direct gather on planar grids.
// ---------------------------------------------------------------------------
struct __attribute__((aligned(4))) F2 { float lo, hi; };
__device__ __forceinline__ F2 ld2(const float* __restrict__ g, int off) {
    return *reinterpret_cast<const F2*>(g + off);
}

__global__ __launch_bounds__(NPTS) void dvgo_render_direct_kernel(
    const float* __restrict__ origins,
    const float* __restrict__ directions,
    const float* __restrict__ lengths,
    const float* __restrict__ dgrid,
    const float* __restrict__ cgrid,
    float* __restrict__ out)
{
    const int ray = blockIdx.x;
    const int p   = threadIdx.x;
    const int idx = ray * NPTS + p;

    const float ox = origins[3 * ray + 0], oy = origins[3 * ray + 1], oz = origins[3 * ray + 2];
    const float dx = directions[3 * ray + 0], dy = directions[3 * ray + 1], dz = directions[3 * ray + 2];
    const float interval = sqrtf(fmaf(dx, dx, fmaf(dy, dy, dz * dz)));
    const float t = __builtin_nontemporal_load(lengths + idx);

    const float S = 0.5f * (float)(GS - 1);
    const float fx = (fmaf(dx, t, ox) + 1.0f) * S;
    const float fy = (fmaf(dy, t, oy) + 1.0f) * S;
    const float fz = (fmaf(dz, t, oz) + 1.0f) * S;

    const float flx = floorf(fx), fly = floorf(fy), flz = floorf(fz);
    const float wx1 = fx - flx, wy1 = fy - fly, wz1 = fz - flz;
    const float wx0 = 1.0f - wx1, wy0 = 1.0f - wy1, wz0 = 1.0f - wz1;

    const int x0 = min(max((int)flx, 0), GS - 1);
    const int y0 = min(max((int)fly, 0), GS - 1);
    const int z0 = min(max((int)flz, 0), GS - 1);
    const int y1 = min(y0 + 1, GS - 1);
    const int z1 = min(z0 + 1, GS - 1);
    const int  xb  = min(x0, GS - 2);
    const bool xhi = (x0 != xb);
    const float wxa = xhi ? 0.0f : wx0;
    const float wxb = xhi ? (wx0 + wx1) : wx1;

    const int o00 = (z0 * GS + y0) * GS + xb;
    const int o01 = (z0 * GS + y1) * GS + xb;
    const int o10 = (z1 * GS + y0) * GS + xb;
    const int o11 = (z1 * GS + y1) * GS + xb;

    const float* __restrict__ cr = cgrid;
    const float* __restrict__ cg = cgrid + GVOL;
    const float* __restrict__ cb = cgrid + 2 * GVOL;
    const F2 d00 = ld2(dgrid, o00), d01 = ld2(dgrid, o01), d10 = ld2(dgrid, o10), d11 = ld2(dgrid, o11);
    const F2 r00 = ld2(cr, o00), r01 = ld2(cr, o01), r10 = ld2(cr, o10), r11 = ld2(cr, o11);
    const F2 g00 = ld2(cg, o00), g01 = ld2(cg, o01), g10 = ld2(cg, o10), g11 = ld2(cg, o11);
    const F2 b00 = ld2(cb, o00), b01 = ld2(cb, o01), b10 = ld2(cb, o10), b11 = ld2(cb, o11);

    auto tri = [&](F2 v00, F2 v01, F2 v10, F2 v11) -> float {
        const float l00 = fmaf(v00.lo, wxa, v00.hi * wxb);
        const float l01 = fmaf(v01.lo, wxa, v01.hi * wxb);
        const float l10 = fmaf(v10.lo, wxa, v10.hi * wxb);
        const float l11 = fmaf(v11.lo, wxa, v11.hi * wxb);
        const float q0 = fmaf(l00, wy0, l01 * wy1);
        const float q1 = fmaf(l10, wy0, l11 * wy1);
        return fmaf(q0, wz0, q1 * wz1);
    };

    const float raw_d = tri(d00, d01, d10, d11);
    const float raw_r = tri(r00, r01, r10, r11);
    const float raw_g = tri(g00, g01, g10, g11);
    const float raw_b = tri(b00, b01, b10, b11);

    const float e    = __builtin_amdgcn_exp2f(fmaf(raw_d, LOG2E, ACT_SHIFT * LOG2E));
    const float dens = 1.0f - __builtin_amdgcn_exp2f(-interval * __builtin_amdgcn_logf(1.0f + e));
    auto sigmoid = [](float x) -> float {
        return __builtin_amdgcn_rcpf(1.0f + __builtin_amdgcn_exp2f(-x * LOG2E));
    };

    __builtin_nontemporal_store(dens, out + idx);
    float* __restrict__ cout = out + (size_t)NRAYS * NPTS + 3 * (size_t)idx;
    __builtin_nontemporal_store(sigmoid(raw_r), cout + 0);
    __builtin_nontemporal_store(sigmoid(raw_g), cout + 1);
    __builtin_nontemporal_store(sigmoid(raw_b), cout + 2);
}

extern "C" void kernel_launch(void* const* d_in, const int* in_sizes, int n_in,
                              void* d_out, int out_size, void* d_ws, size_t ws_size,
                              hipStream_t stream) {
    (void)in_sizes; (void)n_in; (void)out_size;
    const float* origins    = (const float*)d_in[0];
    const float* directions = (const float*)d_in[1];
    const float* lengths    = (const float*)d_in[2];
    const float* dgrid      = (const float*)d_in[3];
    const float* cgrid      = (const float*)d_in[4];
    float* out = (float*)d_out;

    const size_t packed_bytes = (size_t)GVOL * NCH * sizeof(float);  // 65,536,000 B
    if (ws_size >= packed_bytes) {
        float4* packed = (float4*)d_ws;
        dvgo_repack_kernel<<<GVOL / 256, 256, 0, stream>>>(dgrid, cgrid, packed);
        dvgo_render_packed_kernel<<<NRAYS, NPTS, 0, stream>>>(
            origins, directions, lengths, packed, out);
    } else {
        dvgo_render_direct_kernel<<<NRAYS, NPTS, 0, stream>>>(
            origins, directions, lengths, dgrid, cgrid, out);
    }
}